// CostVolumn_90305982366045
// MI455X (gfx1250) — compile-verified
//
#include <hip/hip_runtime.h>

// CDNA5 / gfx1250 cost-volume kernel using V_WMMA_F32_16X16X4_F32.
// All shapes compile-time constants (fixed by the reference):
// B=8, C=192, H=128, W=160, 81 offsets.

typedef __attribute__((ext_vector_type(2))) float v2f;
typedef __attribute__((ext_vector_type(8))) float v8f;

#define SR      4
#define MAXOFF  9                 // 2*SR+1
#define NOFF    81                // MAXOFF*MAXOFF
#define CCH     192
#define KSTEPS  (CCH / 4)         // 48 chained k=4 WMMAs
#define BB      8
#define HH      128
#define WW      160
#define HWC     (HH * WW)         // 20480 (channel stride, elements)
#define SEGS_PER_ROW (WW / 16)    // 10
#define WAVES_PER_BLOCK 8
#define LDS_STRIDE 33             // 32 cols + 1 pad (bank-conflict avoidance)

__device__ __forceinline__ int iclampi(int v, int lo, int hi) {
    return v < lo ? lo : (v > hi ? hi : v);
}

__global__ __launch_bounds__(256) void costvol_wmma_f32(
    const float* __restrict__ c1,
    const float* __restrict__ warped,
    const float* __restrict__ alpha_p,
    float* __restrict__ out)
{
    __shared__ float smem[WAVES_PER_BLOCK * 16 * LDS_STRIDE];

    const int tid  = threadIdx.x;
    const int lane = tid & 31;
    const int wave = tid >> 5;

    // Grid is exact: 8*128*10 segments == gridDim.x * WAVES_PER_BLOCK.
    const int seg = blockIdx.x * WAVES_PER_BLOCK + wave;
    const int b   = seg / (HH * SEGS_PER_ROW);
    int rem       = seg - b * (HH * SEGS_PER_ROW);
    const int h   = rem / SEGS_PER_ROW;
    const int w0  = (rem - h * SEGS_PER_ROW) << 4;

    const int half = lane >> 4;               // 0: K pair {0,1}, 1: {2,3}
    const int m16  = lane & 15;               // pixel (M) / column (N) index

    const float alpha = alpha_p[0];
    const float inv_c = 1.0f / (float)CCH;

    // ---------------- A fragments: c1[b, :, h, w0+m16], register-resident ----
    // VGPR layout (16x4 fp32 A): lane(half,m): v.x = K=2*half, v.y = K=2*half+1
    const float* aBase = c1 + (size_t)b * CCH * HWC
                            + (2 * half) * HWC
                            + h * WW + w0 + m16;
    float a_x[KSTEPS], a_y[KSTEPS];
#pragma unroll
    for (int ks = 0; ks < KSTEPS; ++ks) {
        a_x[ks] = aBase[ks * 4 * HWC];
        a_y[ks] = aBase[ks * 4 * HWC + HWC];
    }

    // ---------------- B column setup (independent of dy) ---------------------
    // Tile0 covers warped columns w0-4 .. w0+11, tile1 covers w0+12 .. w0+27.
    // Pad-masking is deferred to the accumulators: column n of C/D lives
    // in lanes n and n+16, so one per-lane scale after the k-loop suffices.
    const int col0   = w0 - SR + m16;
    const int col1   = col0 + 16;
    const float cm0  = (col0 >= 0 && col0 < WW) ? 1.0f : 0.0f;
    const float cm1  = (col1 < WW) ? 1.0f : 0.0f;   // col1 >= 12 always
    const int col0c  = iclampi(col0, 0, WW - 1);
    const int col1c  = iclampi(col1, 0, WW - 1);

    const float* wBat = warped + (size_t)b * CCH * HWC + (2 * half) * HWC;
    float* smw = smem + wave * 16 * LDS_STRIDE;

#pragma unroll 1
    for (int dy = 0; dy < MAXOFF; ++dy) {
        const int  row  = h + dy - SR;
        const bool rOk  = (row >= 0) && (row < HH);
        const int  rowc = iclampi(row, 0, HH - 1);

        // Clamped (always-valid) base addresses; fixed bases + constant
        // offsets let the compiler fold channel strides into 24-bit imms.
        const float* p0 = wBat + rowc * WW + col0c;
        const float* p1 = wBat + rowc * WW + col1c;

        // Fused pad-mask * mean-scale, applied to accumulators at staging.
        const float s0 = (rOk ? cm0 : 0.0f) * inv_c;
        const float s1 = (rOk ? cm1 : 0.0f) * inv_c;

        // Prefetch next dy's row into cache (global_prefetch_b8).
        if (dy + 1 < MAXOFF) {
            const int rn = iclampi(h + dy + 1 - SR, 0, HH - 1);
            __builtin_prefetch(wBat + rn * WW + col0c, 0, 1);
        }

        v8f acc0 = {};
        v8f acc1 = {};
#pragma unroll
        for (int ks = 0; ks < KSTEPS; ++ks) {
            const int off = ks * 4 * HWC;
            v2f A;  A.x  = a_x[ks];       A.y  = a_y[ks];
            v2f B0; B0.x = p0[off];       B0.y = p0[off + HWC];
            v2f B1; B1.x = p1[off];       B1.y = p1[off + HWC];
            acc0 = __builtin_amdgcn_wmma_f32_16x16x4_f32(
                       false, A, false, B0, (short)0, acc0, false, false);
            acc1 = __builtin_amdgcn_wmma_f32_16x16x4_f32(
                       false, A, false, B1, (short)0, acc1, false, false);
        }

        // ---- stage masked+scaled accumulators to LDS (per-wave region) -----
        // C/D layout: VGPR i, lanes0-15: (M=i, N=lane); lanes16-31: (M=8+i, N=lane-16)
#pragma unroll
        for (int i = 0; i < 8; ++i) {
            const int M = i + half * 8;
            smw[M * LDS_STRIDE + m16]      = acc0[i] * s0;
            smw[M * LDS_STRIDE + 16 + m16] = acc1[i] * s1;
        }
        asm volatile("s_wait_dscnt 0x0" ::: "memory");

        // ---- gather diagonal band, leaky-ReLU, store ------------------------
        // lanes 0-15 handle (m=m16, d=dbase); lanes 16-31 handle d=dbase+1
#pragma unroll
        for (int dbase = 0; dbase < MAXOFF + 1; dbase += 2) {
            const int d = dbase + half;
            const int n = m16 + d;                       // warped col = w0+m+d-4
            const float cost = smw[m16 * LDS_STRIDE + n]; // already scaled
            const float res  = (cost >= 0.0f) ? cost : alpha * cost;
            if (d < MAXOFF) {
                const int o = dy * MAXOFF + d;
                out[(((size_t)b * NOFF + o) * HH + h) * WW + w0 + m16] = res;
            }
        }
        // DS ops are in-order per wave; fence keeps compiler from sinking the
        // next iteration's LDS stores above these loads.
        asm volatile("s_wait_dscnt 0x0" ::: "memory");
    }
}

extern "C" void kernel_launch(void* const* d_in, const int* in_sizes, int n_in,
                              void* d_out, int out_size, void* d_ws, size_t ws_size,
                              hipStream_t stream) {
    (void)in_sizes; (void)n_in; (void)d_ws; (void)ws_size; (void)out_size;

    const float* c1     = (const float*)d_in[0];
    const float* warped = (const float*)d_in[1];
    const float* alpha  = (const float*)d_in[2];
    float*       out    = (float*)d_out;

    const int totalSegs = BB * HH * SEGS_PER_ROW;           // 10240 wave-segments
    const int blocks    = totalSegs / WAVES_PER_BLOCK;      // 1280

    costvol_wmma_f32<<<blocks, 32 * WAVES_PER_BLOCK, 0, stream>>>(
        c1, warped, alpha, out);
}